// EfficientMemorySiLU_65077344469255
// MI455X (gfx1250) — compile-verified
//
#include <hip/hip_runtime.h>
#include <cstdint>

typedef __attribute__((ext_vector_type(2))) float v2f;
typedef __attribute__((ext_vector_type(8))) float v8f;

#define NBINS 4096
#define RANK  16

__device__ __forceinline__ float subval(float v, float thr) {
  // x_sub = where(|x| > thr, 0, x)
  return (__builtin_fabsf(v) > thr) ? 0.0f : v;
}

// ---------------------------------------------------------------- SiLU
__global__ void k_silu(const float* __restrict__ x, float* __restrict__ out, int n4) {
  int i = blockIdx.x * blockDim.x + threadIdx.x;
  int stride = gridDim.x * blockDim.x;
  const float4* xv = (const float4*)x;
  float4* ov = (float4*)out;
  for (; i < n4; i += stride) {
    float4 v = xv[i];
    float4 r;
    r.x = v.x / (1.0f + __expf(-v.x));
    r.y = v.y / (1.0f + __expf(-v.y));
    r.z = v.z / (1.0f + __expf(-v.z));
    r.w = v.w / (1.0f + __expf(-v.w));
    ov[i] = r;
  }
}

// ---------------------------------------------------------------- zero helpers
__global__ void k_zero_u32(unsigned* p, int n) {
  int i = blockIdx.x * blockDim.x + threadIdx.x;
  if (i < n) p[i] = 0u;
}
__global__ void k_zero_f32(float* p, int n) {
  int i = blockIdx.x * blockDim.x + threadIdx.x;
  if (i < n) p[i] = 0.0f;
}

// ---------------------------------------------------------------- abs-max (for histogram range)
__global__ void k_absmax(const float* __restrict__ x, unsigned* maxbits, int n) {
  __shared__ float red[256];
  float m = 0.0f;
  for (int i = blockIdx.x * blockDim.x + threadIdx.x; i < n; i += gridDim.x * blockDim.x)
    m = fmaxf(m, __builtin_fabsf(x[i]));
  red[threadIdx.x] = m;
  __syncthreads();
  for (int s = blockDim.x >> 1; s > 0; s >>= 1) {
    if ((int)threadIdx.x < s) red[threadIdx.x] = fmaxf(red[threadIdx.x], red[threadIdx.x + s]);
    __syncthreads();
  }
  if (threadIdx.x == 0) atomicMax(maxbits, __float_as_uint(red[0]));  // |x| >= 0 -> bit order == float order
}

// ---------------------------------------------------------------- histogram of |x| (LDS-privatized)
__global__ void k_hist(const float* __restrict__ x, const unsigned* __restrict__ maxbits,
                       unsigned* __restrict__ hist, int n) {
  __shared__ unsigned lh[NBINS];
  for (int i = threadIdx.x; i < NBINS; i += blockDim.x) lh[i] = 0u;
  __syncthreads();
  float mx = __uint_as_float(*maxbits);
  float scale = (mx > 0.0f) ? ((float)NBINS / mx) : 0.0f;
  for (int i = blockIdx.x * blockDim.x + threadIdx.x; i < n; i += gridDim.x * blockDim.x) {
    int b = (int)(__builtin_fabsf(x[i]) * scale);
    b = (b > NBINS - 1) ? NBINS - 1 : b;
    atomicAdd(&lh[b], 1u);
  }
  __syncthreads();
  for (int i = threadIdx.x; i < NBINS; i += blockDim.x)
    if (lh[i]) atomicAdd(&hist[i], lh[i]);
}

// ---------------------------------------------------------------- quantile pick (single lane)
__global__ void k_quantile(const unsigned* __restrict__ hist, const unsigned* __restrict__ maxbits,
                           float* thr, float* out_thr, int n, float keep_ratio) {
  if (threadIdx.x != 0 || blockIdx.x != 0) return;
  unsigned long long target = (unsigned long long)((double)keep_ratio * (double)n);
  unsigned long long c = 0;
  int b = 0;
  for (; b < NBINS; ++b) {
    c += hist[b];
    if (c >= target) break;
  }
  float mx = __uint_as_float(*maxbits);
  float t = ((float)(b + 1)) * mx * (1.0f / (float)NBINS);
  *thr = t;
  *out_thr = t;
}

// ---------------------------------------------------------------- deterministic gaussian Omega
__device__ __forceinline__ unsigned pcg_hash(unsigned s) {
  s = s * 747796405u + 2891336453u;
  unsigned w = ((s >> ((s >> 28) + 4u)) ^ s) * 277803737u;
  return (w >> 22) ^ w;
}
__global__ void k_omega(float* __restrict__ omega, int n) {
  int i = blockIdx.x * blockDim.x + threadIdx.x;
  if (i >= n) return;
  unsigned a = pcg_hash(2u * (unsigned)i + 1u);
  unsigned b = pcg_hash(2u * (unsigned)i + 2u);
  float u1 = ((float)a + 1.0f) * (1.0f / 4294967296.0f);
  float u2 = (float)b * (1.0f / 4294967296.0f);
  omega[i] = sqrtf(-2.0f * __logf(u1)) * __cosf(6.28318530718f * u2);
}

// ================================================================ WMMA GEMMs
// All use V_WMMA_F32_16X16X4_F32: one wave owns a 16x16 f32 tile, K advances by 4.
// A 16x4 lane layout: lanes 0-15 hold K=0,1 (v0,v1); lanes 16-31 hold K=2,3.
// B 4x16 mirrored; C/D: VGPR j -> M=j (lanes 0-15) / M=j+8 (lanes 16-31), N=lane&15.

// ---- main GEMM: C[Mrows,16] = mask(x)[Mrows,K] @ W[K,16]
// W panel is shared by all 8 waves of the block -> stage K-chunks of W into LDS
// with double-buffered GLOBAL_LOAD_ASYNC_TO_LDS_B128 (ASYNCcnt-tracked).
// Requires: Mrows % 128 == 0, K % WCH == 0, blockDim == 256.
#define WCH 128
__global__ void k_gemm_x_w16(const float* __restrict__ x, const float* __restrict__ thrp,
                             const float* __restrict__ W, float* __restrict__ C,
                             int Mrows, int K) {
  __shared__ float wbuf[2][WCH * 16];
  const int tid  = threadIdx.x;
  const int lane = tid & 31;
  const int wave = tid >> 5;
  const int tile = blockIdx.x * (blockDim.x >> 5) + wave;
  const float thr = *thrp;
  const int m0 = tile * 16;
  const int half = lane >> 4;
  const int l = lane & 15;
  const float* arow = x + (size_t)(m0 + l) * K + half * 2;

  // issue async copy of one W chunk (WCH x 16 floats) into wbuf[buf]
  auto stage = [&](int buf, int kbase) {
    const float* gsrc = W + (size_t)kbase * 16;
    // 2048 floats / 4 per B128 = 512 issues; 256 threads -> exactly 2 per thread
    for (int i = tid * 4; i < WCH * 16; i += blockDim.x * 4) {
      unsigned ldsoff = (unsigned)(uintptr_t)(&wbuf[buf][i]);
      unsigned long long ga = (unsigned long long)(uintptr_t)(gsrc + i);
      asm volatile("global_load_async_to_lds_b128 %0, %1, off"
                   :: "v"(ldsoff), "v"(ga) : "memory");
    }
  };

  v8f acc = {};
  stage(0, 0);
  int buf = 0;
  for (int kc = 0; kc < K; kc += WCH) {
    if (kc + WCH < K) {
      stage(buf ^ 1, kc + WCH);                       // prefetch next chunk
      asm volatile("s_wait_asynccnt 0x2" ::: "memory");  // retire only the older chunk
    } else {
      asm volatile("s_wait_asynccnt 0x0" ::: "memory");
    }
    __syncthreads();
    __builtin_prefetch(arow + kc + WCH, 0, 1);
    const float* wb = &wbuf[buf][0];
#pragma unroll 4
    for (int kk = 0; kk < WCH; kk += 4) {
      float2 av = *(const float2*)(arow + kc + kk);
      v2f a;
      a.x = subval(av.x, thr);
      a.y = subval(av.y, thr);
      v2f b;
      b.x = wb[(kk + half * 2 + 0) * 16 + l];
      b.y = wb[(kk + half * 2 + 1) * 16 + l];
      acc = __builtin_amdgcn_wmma_f32_16x16x4_f32(false, a, false, b, (short)0, acc, false, false);
    }
    __syncthreads();
    buf ^= 1;
  }
#pragma unroll
  for (int j = 0; j < 8; ++j)
    C[(size_t)(m0 + j + half * 8) * 16 + l] = acc[j];
}

// C[D,16] = mask(x).T[D,K] @ W[K,16]   (x is [K,D] row-major)
__global__ void k_gemm_xt_w16(const float* __restrict__ x, const float* __restrict__ thrp,
                              const float* __restrict__ W, float* __restrict__ C,
                              int D, int K) {
  const int lane = threadIdx.x & 31;
  const int wave = threadIdx.x >> 5;
  const int tile = blockIdx.x * (blockDim.x >> 5) + wave;
  if (tile * 16 >= D) return;
  const float thr = *thrp;
  const int d0 = tile * 16;
  const int half = lane >> 4;
  const int l = lane & 15;
  v8f acc = {};
  for (int k = 0; k < K; k += 4) {
    int k0 = k + half * 2;
    v2f a;
    a.x = subval(x[(size_t)(k0 + 0) * D + d0 + l], thr);
    a.y = subval(x[(size_t)(k0 + 1) * D + d0 + l], thr);
    v2f b;
    b.x = W[(k0 + 0) * 16 + l];
    b.y = W[(k0 + 1) * 16 + l];
    acc = __builtin_amdgcn_wmma_f32_16x16x4_f32(false, a, false, b, (short)0, acc, false, false);
  }
#pragma unroll
  for (int j = 0; j < 8; ++j)
    C[(size_t)(d0 + j + half * 8) * 16 + l] = acc[j];
}

// Bout[16,D] = Q.T[16,K] @ mask(x)[K,D]   (Q is [K,16] row-major)
__global__ void k_gemm_qt_x(const float* __restrict__ x, const float* __restrict__ thrp,
                            const float* __restrict__ Q, float* __restrict__ Bout,
                            int D, int K) {
  const int lane = threadIdx.x & 31;
  const int wave = threadIdx.x >> 5;
  const int tile = blockIdx.x * (blockDim.x >> 5) + wave;
  if (tile * 16 >= D) return;
  const float thr = *thrp;
  const int n0 = tile * 16;
  const int half = lane >> 4;
  const int l = lane & 15;
  v8f acc = {};
  for (int k = 0; k < K; k += 4) {
    int k0 = k + half * 2;
    v2f a;
    a.x = Q[(k0 + 0) * 16 + l];
    a.y = Q[(k0 + 1) * 16 + l];
    v2f b;
    b.x = subval(x[(size_t)(k0 + 0) * D + n0 + l], thr);
    b.y = subval(x[(size_t)(k0 + 1) * D + n0 + l], thr);
    acc = __builtin_amdgcn_wmma_f32_16x16x4_f32(false, a, false, b, (short)0, acc, false, false);
  }
#pragma unroll
  for (int j = 0; j < 8; ++j)
    Bout[(size_t)(j + half * 8) * D + n0 + l] = acc[j];
}

// Q[Mrows,16] = M[Mrows,16] @ T[16,16]   (K = 16: four WMMA steps)
__global__ void k_gemm_m_t16(const float* __restrict__ M, const float* __restrict__ T,
                             float* __restrict__ Q, int Mrows) {
  const int lane = threadIdx.x & 31;
  const int wave = threadIdx.x >> 5;
  const int tile = blockIdx.x * (blockDim.x >> 5) + wave;
  if (tile * 16 >= Mrows) return;
  const int m0 = tile * 16;
  const int half = lane >> 4;
  const int l = lane & 15;
  v8f acc = {};
#pragma unroll
  for (int k = 0; k < 16; k += 4) {
    int k0 = k + half * 2;
    v2f a;
    a.x = M[(size_t)(m0 + l) * 16 + k0 + 0];
    a.y = M[(size_t)(m0 + l) * 16 + k0 + 1];
    v2f b;
    b.x = T[(k0 + 0) * 16 + l];
    b.y = T[(k0 + 1) * 16 + l];
    acc = __builtin_amdgcn_wmma_f32_16x16x4_f32(false, a, false, b, (short)0, acc, false, false);
  }
#pragma unroll
  for (int j = 0; j < 8; ++j)
    Q[(size_t)(m0 + j + half * 8) * 16 + l] = acc[j];
}

// G[16,16] += M.T @ M, chunked over K with f32 atomics (G pre-zeroed)
#define GRAM_CHUNK 512
__global__ void k_gram(const float* __restrict__ M, float* __restrict__ G, int Mrows) {
  const int lane = threadIdx.x & 31;
  const int wave = threadIdx.x >> 5;
  const int chunk = blockIdx.x * (blockDim.x >> 5) + wave;
  const int k0 = chunk * GRAM_CHUNK;
  if (k0 >= Mrows) return;
  const int kend = (k0 + GRAM_CHUNK < Mrows) ? (k0 + GRAM_CHUNK) : Mrows;
  const int half = lane >> 4;
  const int l = lane & 15;
  v8f acc = {};
  for (int k = k0; k < kend; k += 4) {
    int kk = k + half * 2;
    v2f a;
    a.x = M[(size_t)(kk + 0) * 16 + l];
    a.y = M[(size_t)(kk + 1) * 16 + l];
    // G = M^T M is symmetric: B operand carries the same lane data as A
    acc = __builtin_amdgcn_wmma_f32_16x16x4_f32(false, a, false, a, (short)0, acc, false, false);
  }
#pragma unroll
  for (int j = 0; j < 8; ++j)
    atomicAdd(&G[(j + half * 8) * 16 + l], acc[j]);
}

// ---------------------------------------------------------------- Cholesky (G = U^T U) + invert U (upper-tri)
__global__ void k_cholinv(const float* __restrict__ G, float* __restrict__ Tinv) {
  if (threadIdx.x != 0 || blockIdx.x != 0) return;
  float U[RANK][RANK];
  for (int i = 0; i < RANK; ++i)
    for (int j = 0; j < RANK; ++j) U[i][j] = 0.0f;
  for (int i = 0; i < RANK; ++i) {
    float s = G[i * RANK + i];
    for (int k = 0; k < i; ++k) s -= U[k][i] * U[k][i];
    float d = sqrtf(fmaxf(s, 1e-20f));
    U[i][i] = d;
    float inv_d = 1.0f / d;
    for (int j = i + 1; j < RANK; ++j) {
      float t = G[i * RANK + j];
      for (int k = 0; k < i; ++k) t -= U[k][i] * U[k][j];
      U[i][j] = t * inv_d;
    }
  }
  float Ti[RANK][RANK];
  for (int i = 0; i < RANK; ++i)
    for (int j = 0; j < RANK; ++j) Ti[i][j] = 0.0f;
  for (int j = 0; j < RANK; ++j) {
    Ti[j][j] = 1.0f / U[j][j];
    for (int i = j - 1; i >= 0; --i) {
      float s = 0.0f;
      for (int k = i + 1; k <= j; ++k) s += U[i][k] * Ti[k][j];
      Ti[i][j] = -s / U[i][i];
    }
  }
  for (int i = 0; i < RANK; ++i)
    for (int j = 0; j < RANK; ++j) Tinv[i * RANK + j] = Ti[i][j];
}

// ---------------------------------------------------------------- W = B B^T  (16x16 from [16,D])
__global__ void k_bbt(const float* __restrict__ B, float* __restrict__ W, int D) {
  int i = threadIdx.x >> 4;
  int j = threadIdx.x & 15;
  const float* bi = B + (size_t)i * D;
  const float* bj = B + (size_t)j * D;
  float s = 0.0f;
  for (int d = 0; d < D; ++d) s = fmaf(bi[d], bj[d], s);
  W[i * RANK + j] = s;
}

// ---------------------------------------------------------------- Jacobi eigensolver (16x16 symmetric)
__global__ void k_jacobi(const float* __restrict__ Win, float* __restrict__ Ub, float* __restrict__ S) {
  if (threadIdx.x != 0 || blockIdx.x != 0) return;
  float A[RANK][RANK], V[RANK][RANK];
  for (int i = 0; i < RANK; ++i)
    for (int j = 0; j < RANK; ++j) {
      A[i][j] = Win[i * RANK + j];
      V[i][j] = (i == j) ? 1.0f : 0.0f;
    }
  for (int sweep = 0; sweep < 12; ++sweep) {
    for (int p = 0; p < RANK - 1; ++p) {
      for (int q = p + 1; q < RANK; ++q) {
        float apq = A[p][q];
        if (__builtin_fabsf(apq) < 1e-12f) continue;
        float app = A[p][p], aqq = A[q][q];
        float theta = (aqq - app) / (2.0f * apq);
        float t = ((theta >= 0.0f) ? 1.0f : -1.0f) /
                  (__builtin_fabsf(theta) + sqrtf(theta * theta + 1.0f));
        float c = rsqrtf(t * t + 1.0f);
        float s = t * c;
        float npp = c * c * app - 2.0f * s * c * apq + s * s * aqq;
        float nqq = s * s * app + 2.0f * s * c * apq + c * c * aqq;
        for (int k = 0; k < RANK; ++k) {
          if (k == p || k == q) continue;
          float akp = A[k][p], akq = A[k][q];
          A[k][p] = A[p][k] = c * akp - s * akq;
          A[k][q] = A[q][k] = s * akp + c * akq;
        }
        A[p][p] = npp;
        A[q][q] = nqq;
        A[p][q] = A[q][p] = 0.0f;
        for (int k = 0; k < RANK; ++k) {
          float vkp = V[k][p], vkq = V[k][q];
          V[k][p] = c * vkp - s * vkq;
          V[k][q] = s * vkp + c * vkq;
        }
      }
    }
  }
  int idx[RANK];
  float ev[RANK];
  for (int i = 0; i < RANK; ++i) { idx[i] = i; ev[i] = A[i][i]; }
  for (int i = 0; i < RANK - 1; ++i)
    for (int j = i + 1; j < RANK; ++j)
      if (ev[idx[j]] > ev[idx[i]]) { int tt = idx[i]; idx[i] = idx[j]; idx[j] = tt; }
  for (int i = 0; i < RANK; ++i) {
    S[i] = sqrtf(fmaxf(ev[idx[i]], 0.0f));
    for (int k = 0; k < RANK; ++k) Ub[k * RANK + i] = V[k][idx[i]];
  }
}

// ---------------------------------------------------------------- Vh = S^-1 Ub^T B ; R = Vh^T ; Rinv = Vh
__global__ void k_vh(const float* __restrict__ B, const float* __restrict__ Ub,
                     const float* __restrict__ S, float* __restrict__ R,
                     float* __restrict__ Rinv, int D) {
  int d = blockIdx.x * blockDim.x + threadIdx.x;
  if (d >= D) return;
  float bcol[RANK];
#pragma unroll
  for (int k = 0; k < RANK; ++k) bcol[k] = B[(size_t)k * D + d];
#pragma unroll
  for (int r = 0; r < RANK; ++r) {
    float s = 0.0f;
#pragma unroll
    for (int k = 0; k < RANK; ++k) s = fmaf(Ub[k * RANK + r], bcol[k], s);
    float sv = S[r];
    float v = (sv > 1e-12f) ? (s / sv) : 0.0f;
    Rinv[(size_t)r * D + d] = v;
    R[(size_t)d * RANK + r] = v;
  }
}

// ================================================================ launcher
extern "C" void kernel_launch(void* const* d_in, const int* in_sizes, int n_in,
                              void* d_out, int out_size, void* d_ws, size_t ws_size,
                              hipStream_t stream) {
  const float* x = (const float*)d_in[0];
  const int n = in_sizes[0];      // 4*2048*4096 = 33554432
  const int D = 4096;
  const int Nr = n / D;           // 8192

  float* out = (float*)d_out;
  float* out_thr = out + n;
  float* Rout = out + n + 1;                       // [D,16]
  float* Rinv_out = Rout + (size_t)D * RANK;       // [16,D]

  // workspace carve (256B aligned)
  uintptr_t p = (uintptr_t)d_ws;
  auto carve = [&](size_t bytes) -> void* {
    uintptr_t q = (p + 255) & ~(uintptr_t)255;
    p = q + bytes;
    return (void*)q;
  };
  unsigned* maxbits = (unsigned*)carve(4);
  unsigned* hist    = (unsigned*)carve(NBINS * sizeof(unsigned));
  float* thr   = (float*)carve(4);
  float* Omega = (float*)carve((size_t)D * RANK * 4);
  float* M1    = (float*)carve((size_t)Nr * RANK * 4);
  float* Qn    = (float*)carve((size_t)Nr * RANK * 4);
  float* M2    = (float*)carve((size_t)D * RANK * 4);
  float* Qd    = (float*)carve((size_t)D * RANK * 4);
  float* G     = (float*)carve(RANK * RANK * 4);
  float* Tinv  = (float*)carve(RANK * RANK * 4);
  float* Bm    = (float*)carve((size_t)RANK * D * 4);
  float* W     = (float*)carve(RANK * RANK * 4);
  float* Ub    = (float*)carve(RANK * RANK * 4);
  float* S     = (float*)carve(RANK * 4);
  (void)ws_size; (void)n_in; (void)out_size;

  // 1) SiLU (independent of the rest)
  k_silu<<<2048, 256, 0, stream>>>(x, out, n / 4);

  // 2) quantile threshold via max + histogram
  k_zero_u32<<<1, 256, 0, stream>>>(maxbits, 1);
  k_zero_u32<<<(NBINS + 255) / 256, 256, 0, stream>>>(hist, NBINS);
  k_absmax<<<2048, 256, 0, stream>>>(x, maxbits, n);
  k_hist<<<2048, 256, 0, stream>>>(x, maxbits, hist, n);
  k_quantile<<<1, 32, 0, stream>>>(hist, maxbits, thr, out_thr, n, 0.99f);

  // 3) randomized SVD pipeline (all GEMMs mask x inline with thr)
  k_omega<<<(D * RANK + 255) / 256, 256, 0, stream>>>(Omega, D * RANK);

  auto qr = [&](const float* Min, float* Qout, int rows) {
    k_zero_f32<<<1, 256, 0, stream>>>(G, RANK * RANK);
    int waves = rows / GRAM_CHUNK;
    k_gram<<<(waves + 7) / 8, 256, 0, stream>>>(Min, G, rows);
    k_cholinv<<<1, 32, 0, stream>>>(G, Tinv);
    int tiles = rows / 16;
    k_gemm_m_t16<<<(tiles + 7) / 8, 256, 0, stream>>>(Min, Tinv, Qout, rows);
  };

  // Q = qr(A @ Omega)   (Nr/16 tiles, 8 waves per block -> exact grid, no partial blocks)
  k_gemm_x_w16<<<Nr / 128, 256, 0, stream>>>(x, thr, Omega, M1, Nr, D);
  qr(M1, Qn, Nr);

  // niter = 2 power iterations
  for (int it = 0; it < 2; ++it) {
    k_gemm_xt_w16<<<(D / 16 + 7) / 8, 256, 0, stream>>>(x, thr, Qn, M2, D, Nr);
    qr(M2, Qd, D);
    k_gemm_x_w16<<<Nr / 128, 256, 0, stream>>>(x, thr, Qd, M1, Nr, D);
    qr(M1, Qn, Nr);
  }

  // B = Q^T A ; small SVD of B via eigen of B B^T
  k_gemm_qt_x<<<(D / 16 + 7) / 8, 256, 0, stream>>>(x, thr, Qn, Bm, D, Nr);
  k_bbt<<<1, 256, 0, stream>>>(Bm, W, D);
  k_jacobi<<<1, 32, 0, stream>>>(W, Ub, S);
  k_vh<<<D / 256, 256, 0, stream>>>(Bm, Ub, S, Rout, Rinv_out, D);
}